// Router_15333033246887
// MI455X (gfx1250) — compile-verified
//
#include <hip/hip_runtime.h>

typedef __attribute__((ext_vector_type(2))) float v2f;
typedef __attribute__((ext_vector_type(8))) float v8f;

#define D_DIM 1024
#define E_NUM 16
#define S_LEN 2048
#define B_NUM 4
#define CAP_C 256

// ---------------------------------------------------------------------------
// Kernel 1: gating logits via V_WMMA_F32_16X16X4_F32 (exact fp32, matching
// the reference numerics for the discrete top-k), then softmax + top-2.
// One wave = one 16-token x 16-expert tile; K marches in steps of 4.
// ---------------------------------------------------------------------------
__global__ __launch_bounds__(256) void moe_gate_kernel(
    const float* __restrict__ x,       // [B*S, D]
    const float* __restrict__ w,       // [D, E] row-major
    const float* __restrict__ bias,    // [E]
    int*   __restrict__ idx_out,       // [B*S, 2]
    float* __restrict__ gate_out)      // [B*S, 2]
{
    __shared__ float sm[8][16][17];          // +1 pad: no LDS bank conflicts
    const int lane = threadIdx.x & 31;
    const int wv   = threadIdx.x >> 5;
    const int tile = blockIdx.x * 8 + wv;    // 512 tiles of 16 tokens
    const int row0 = tile * 16;
    const int m    = lane & 15;              // M (A) / N (B) position
    const int hi   = lane >> 4;              // 0: K+0/K+1, 1: K+2/K+3

    const float* arow = x + (size_t)(row0 + m) * D_DIM;

    v8f c = {};
    #pragma unroll 4
    for (int k0 = 0; k0 < D_DIM; k0 += 4) {
        const int ka = k0 + hi * 2;
        // A 16x4 f32 fragment: lane m holds A[m][ka], A[m][ka+1]
        v2f a = *(const v2f*)(arow + ka);
        // B 4x16 f32 fragment: lane holds W[ka][m], W[ka+1][m]
        v2f b;
        b.x = w[(size_t)ka * E_NUM + m];
        b.y = w[(size_t)(ka + 1) * E_NUM + m];
        c = __builtin_amdgcn_wmma_f32_16x16x4_f32(
                false, a, false, b, (short)0, c, false, false);
    }

    // C/D layout: VGPR r, lanes 0-15 -> M=r, lanes 16-31 -> M=8+r, N=lane&15
    #pragma unroll
    for (int r = 0; r < 8; ++r)
        sm[wv][hi * 8 + r][m] = c[r];
    __syncthreads();

    if (lane < 16) {                       // one thread per token: softmax+top2
        float lg[16];
        float mx = -3.4e38f;
        #pragma unroll
        for (int e = 0; e < 16; ++e) {
            lg[e] = sm[wv][lane][e] + bias[e];
            mx = fmaxf(mx, lg[e]);
        }
        float sum = 0.0f;
        #pragma unroll
        for (int e = 0; e < 16; ++e) { lg[e] = __expf(lg[e] - mx); sum += lg[e]; }
        const float inv = 1.0f / sum;

        float v0 = -1.0f, v1 = -1.0f; int i0 = 0, i1 = 0;
        #pragma unroll
        for (int e = 0; e < 16; ++e) {     // strict '>' => ties keep lower index
            const float p = lg[e] * inv;
            if (p > v0)      { v1 = v0; i1 = i0; v0 = p; i0 = e; }
            else if (p > v1) { v1 = p;  i1 = e; }
        }
        const int T = row0 + lane;
        idx_out[T * 2 + 0]  = i0;
        idx_out[T * 2 + 1]  = i1;
        gate_out[T * 2 + 0] = v0;
        gate_out[T * 2 + 1] = v1;
    }
}

// ---------------------------------------------------------------------------
// Kernel 2: per-(b,k) sequential position-in-expert scan over S (ballot-based
// intra-wave ranking, replicated per-expert counters), then sparse scatter of
// gate values into combine/dispatch. 8 waves total; trivially cheap.
// ---------------------------------------------------------------------------
__global__ __launch_bounds__(32) void moe_scan_scatter_kernel(
    const int*   __restrict__ idx_in,   // [B*S, 2]
    const float* __restrict__ gate_in,  // [B*S, 2]
    const int*   __restrict__ cap_ptr,  // scalar expert_capacity
    float* __restrict__ combine,        // [B,S,E,C-1]
    float* __restrict__ dispatch)       // [B,S,E,C-1]
{
    const int b    = blockIdx.x >> 1;
    const int k    = blockIdx.x & 1;
    const int lane = threadIdx.x;
    const int cap  = cap_ptr[0];
    const unsigned ltmask = (1u << lane) - 1u;

    unsigned counts[16];
    #pragma unroll
    for (int e = 0; e < 16; ++e) counts[e] = 0u;

    for (int ch = 0; ch < S_LEN / 32; ++ch) {
        const int s = ch * 32 + lane;
        const size_t tk = ((size_t)b * S_LEN + s) * 2 + k;
        const int   my_e = idx_in[tk];
        const float g    = gate_in[tk];

        unsigned pos = 0u;
        #pragma unroll
        for (int e = 0; e < 16; ++e) {
            const unsigned msk = __builtin_amdgcn_ballot_w32(my_e == e);
            if (my_e == e) pos = counts[e] + __popc(msk & ltmask) + 1u;
            counts[e] += (unsigned)__popc(msk);
        }

        // valid = position < capacity (1-indexed); output drops slot 0
        if ((int)pos < cap && pos >= 1u && pos <= (unsigned)(CAP_C - 1)) {
            const size_t o = (((size_t)b * S_LEN + s) * E_NUM + my_e)
                             * (CAP_C - 1) + (pos - 1u);
            combine[o]  = g;
            dispatch[o] = 1.0f;
        }
    }
}

extern "C" void kernel_launch(void* const* d_in, const int* in_sizes, int n_in,
                              void* d_out, int out_size, void* d_ws, size_t ws_size,
                              hipStream_t stream) {
    const float* x    = (const float*)d_in[0];
    const float* w    = (const float*)d_in[1];
    const float* bias = (const float*)d_in[2];
    const int*   cap  = (const int*)d_in[3];

    const size_t ntok          = (size_t)B_NUM * S_LEN;               // 8192
    const size_t combine_elems = ntok * E_NUM * (CAP_C - 1);          // 33.4M
    float* combine  = (float*)d_out;
    float* dispatch = combine + combine_elems;

    int*   idx_ws  = (int*)d_ws;
    float* gate_ws = (float*)((char*)d_ws + ntok * 2 * sizeof(int));

    // Output is ~267 MB and nearly all zeros: bulk clear is the dominant cost
    // (~11.5 us at 23.3 TB/s); scatter only touches <=16384 elements after.
    hipMemsetAsync(d_out, 0, (size_t)out_size * sizeof(float), stream);

    moe_gate_kernel<<<64, 256, 0, stream>>>(x, w, bias, idx_ws, gate_ws);
    moe_scan_scatter_kernel<<<8, 32, 0, stream>>>(idx_ws, gate_ws, cap,
                                                  combine, dispatch);
}